// StereoConv_61727269978514
// MI455X (gfx1250) — compile-verified
//
#include <hip/hip_runtime.h>
#include <stdint.h>

// ---------------------------------------------------------------------------
// StereoConv (disparity shift-and-blend):
//   out[b,d,h,w] = w1 * X[b,h, w-(d+1)/2] + w2 * X[b,h, w+d/2+1]   (0 if OOB)
// Output: (2,512,512,512) f32 = 1 GiB; input 2 MiB. 3 FLOP/elem -> pure
// store-bandwidth problem (floor ~46us @ 23.3 TB/s). Strategy:
//   * block = (b, d, 16-row chunk): shifts are scalar per block, output slab
//     contiguous -> coalesced 128-bit non-temporal stores.
//   * stage the 16 reused input rows in LDS via CDNA5 async global->LDS b128
//     copies (ASYNCcnt) instead of per-use global gathers.
// ---------------------------------------------------------------------------

#define W_DIM   512
#define H_DIM   512
#define ROWS    16
#define THREADS 256

typedef float v4f __attribute__((ext_vector_type(4)));
// Exact pointee type expected by __builtin_amdgcn_global_load_async_to_lds_b128:
typedef int v4i_vs __attribute__((vector_size(16)));
typedef __attribute__((address_space(1))) v4i_vs* gptr_b128;
typedef __attribute__((address_space(3))) v4i_vs* lptr_b128;

#if defined(__has_builtin)
#if __has_builtin(__builtin_amdgcn_global_load_async_to_lds_b128)
#define USE_ASYNC_LDS 1
#endif
#endif
#ifndef USE_ASYNC_LDS
#define USE_ASYNC_LDS 0
#endif

// Build an LDS (addrspace(3)) pointer from a generic pointer: the low 32 bits
// of a generic LDS pointer are the DS byte offset on AMDGPU (inttoptr, always
// legal). Global generic pointers are value-identical to addrspace(1).
__device__ __forceinline__ static lptr_b128 lds_cast(void* p) {
  return (lptr_b128)(uint32_t)(uintptr_t)p;
}
__device__ __forceinline__ static gptr_b128 glob_cast(const void* p) {
  return (gptr_b128)(uintptr_t)p;
}

__global__ __launch_bounds__(THREADS)
void stereo_shift_kernel(const float* __restrict__ x,
                         const float* __restrict__ w1p,
                         const float* __restrict__ w2p,
                         float* __restrict__ out)
{
  __shared__ float smem[ROWS * W_DIM];   // 32 KB: 16 input rows

  const int hc  = blockIdx.x;            // row chunk: 0..H/ROWS-1
  const int d   = blockIdx.y;            // disparity: 0..511
  const int b   = blockIdx.z;            // batch
  const int h0  = hc * ROWS;
  const int tid = threadIdx.x;

  const float w1 = w1p[0];               // uniform -> s_load
  const float w2 = w2p[0];
  const int s1 = -((d + 1) >> 1);        // s1 in [-256, 0]
  const int s2 = (d >> 1) + 1;           // s2 in [1, 256]

  // ---- Stage x[b, h0:h0+ROWS, :] into LDS ------------------------------
  const float* src = x + ((size_t)b * H_DIM + (size_t)h0) * W_DIM;

#if USE_ASYNC_LDS
  #pragma unroll
  for (int k = 0; k < (ROWS * W_DIM / 4) / THREADS; ++k) {   // 8 iters
    const int q = tid + k * THREADS;                          // float4 index
    __builtin_amdgcn_global_load_async_to_lds_b128(
        glob_cast(src + q * 4), lds_cast(&smem[q * 4]),
        /*offset=*/0, /*cpol=*/0);
  }
#if defined(__has_builtin) && __has_builtin(__builtin_amdgcn_s_wait_asynccnt)
  __builtin_amdgcn_s_wait_asynccnt(0);
#else
  asm volatile("s_wait_asynccnt 0x0" ::: "memory");
#endif
#else
  #pragma unroll
  for (int k = 0; k < (ROWS * W_DIM / 4) / THREADS; ++k) {
    const int q = tid + k * THREADS;
    ((v4f*)smem)[q] = ((const v4f*)src)[q];
  }
#endif
  __syncthreads();

  // ---- Compute: each thread emits 4 consecutive floats per row-pair ----
  const int rhalf = tid >> 7;            // which of 2 rows this iter
  const int wq    = (tid & 127) << 2;    // starting w (multiple of 4)
  float* obase = out + (((size_t)b * W_DIM + (size_t)d) * H_DIM + (size_t)h0) * W_DIM;

  #pragma unroll
  for (int it = 0; it < ROWS / 2; ++it) {
    const int r = it * 2 + rhalf;
    const float* row = smem + r * W_DIM;
    v4f o;
    #pragma unroll
    for (int j = 0; j < 4; ++j) {
      const int w  = wq + j;
      const int i1 = w + s1;                       // can only underflow
      const int i2 = w + s2;                       // can only overflow
      const int c1 = (i1 < 0) ? 0 : i1;
      const int c2 = (i2 > W_DIM - 1) ? (W_DIM - 1) : i2;
      float v1 = row[c1];
      float v2 = row[c2];
      v1 = (i1 >= 0)    ? v1 : 0.0f;
      v2 = (i2 < W_DIM) ? v2 : 0.0f;
      o[j] = w1 * v1 + w2 * v2;
    }
    // 1 GiB streaming output, never re-read: non-temporal 128-bit store.
    __builtin_nontemporal_store(o, (v4f*)(obase + (size_t)r * W_DIM + wq));
  }
}

extern "C" void kernel_launch(void* const* d_in, const int* in_sizes, int n_in,
                              void* d_out, int out_size, void* d_ws, size_t ws_size,
                              hipStream_t stream) {
  const float* x  = (const float*)d_in[0];
  const float* w1 = (const float*)d_in[1];
  const float* w2 = (const float*)d_in[2];
  float* out = (float*)d_out;

  const int B = in_sizes[0] / (H_DIM * W_DIM);   // C == 1
  dim3 grid(H_DIM / ROWS, W_DIM, B);             // (32, 512, 2)
  stereo_shift_kernel<<<grid, dim3(THREADS), 0, stream>>>(x, w1, w2, out);
}